// ExpectedKernelModel_7017976562076
// MI455X (gfx1250) — compile-verified
//
#include <hip/hip_runtime.h>
#include <hip/hip_bf16.h>
#include <math.h>

// Problem constants (match reference)
#define DLAT 64
#define MUC  512
#define MIC  512
#define BSZ  4096

typedef float v2f __attribute__((ext_vector_type(2)));
typedef float v8f __attribute__((ext_vector_type(8)));

// CDNA5 f32 WMMA: D(16x16) = A(16x4) * B(4x16) + C, fp32 throughout.
__device__ __forceinline__ v8f wmma_f32_k4(v2f a, v2f b, v8f c) {
    // 8 args: (neg_a, A, neg_b, B, c_mod, C, reuse_a, reuse_b)
    return __builtin_amdgcn_wmma_f32_16x16x4_f32(
        false, a, false, b, (short)0, c, false, false);
}

// -------------------- 1) row gathers: m1 = um[uidx], m2 = im[iidx] ----------
__global__ void gather_rows(const int* __restrict__ idx,
                            const float* __restrict__ table,
                            float* __restrict__ out, int ncols) {
    int row = blockIdx.x;
    long long src = idx[row];
    const float4* s = (const float4*)(table + src * (long long)ncols);
    float4*       d = (float4*)(out + (long long)row * ncols);
    for (int c = threadIdx.x; c < (ncols >> 2); c += blockDim.x) d[c] = s[c];
}

// -------------------- 2) gm[i,j] = exp(0.5*(-logdet - D*log2pi - maha)) -----
__global__ void gm_kernel(const float* __restrict__ g1,
                          const float* __restrict__ g2,
                          float* __restrict__ gm) {
    int t = blockIdx.x * blockDim.x + threadIdx.x;   // t in [0, MUC*MIC)
    int i = t / MIC, j = t % MIC;
    const float* mp = g1 + i * 2 * DLAT;
    const float* sp = mp + DLAT;
    const float* mq = g2 + j * 2 * DLAT;
    const float* sq = mq + DLAT;
    float logdet = 0.f, maha = 0.f;
    #pragma unroll 8
    for (int d = 0; d < DLAT; ++d) {
        float s    = sp[d] + sq[d];
        float diff = mp[d] - mq[d];
        logdet += __logf(s);
        maha   += diff * diff * __frcp_rn(s);
    }
    const float dlog2pi = 64.0f * 1.83787706640934548356f; // D * log(2*pi)
    gm[t] = __expf(0.5f * (-logdet - dlog2pi - maha));
}

// -------------------- 3) tp = softmax(-gm) along rows (BETA = 1) ------------
__global__ void softmax_kernel(const float* __restrict__ gm,
                               float* __restrict__ tp) {
    __shared__ float red[256];
    int row = blockIdx.x;
    const float* x = gm + row * MIC;
    float m = -3.4e38f;
    for (int j = threadIdx.x; j < MIC; j += 256) m = fmaxf(m, -x[j]);
    red[threadIdx.x] = m; __syncthreads();
    for (int s = 128; s > 0; s >>= 1) {
        if (threadIdx.x < s) red[threadIdx.x] = fmaxf(red[threadIdx.x], red[threadIdx.x + s]);
        __syncthreads();
    }
    m = red[0]; __syncthreads();
    float sum = 0.f;
    for (int j = threadIdx.x; j < MIC; j += 256) sum += __expf(-x[j] - m);
    red[threadIdx.x] = sum; __syncthreads();
    for (int s = 128; s > 0; s >>= 1) {
        if (threadIdx.x < s) red[threadIdx.x] += red[threadIdx.x + s];
        __syncthreads();
    }
    float inv = __frcp_rn(red[0]);
    for (int j = threadIdx.x; j < MIC; j += 256)
        tp[row * MIC + j] = __expf(-x[j] - m) * inv;
}

// -------------------- NT GEMM via f32 WMMA ----------------------------------
// C[M,N] = A[M,K] @ Bt[N,K]^T.  blockIdx.z selects the (A,B,C) triple.
// Wave tile: 32(M) x 64(N) = 8 accumulators; block = 8 waves stacked on M
// -> block tile 256 x 64.
// TRANS_STORE: write C transposed as Ct[N,M] (row stride M). For a fixed lane
//   the 8 accumulator elements are M-contiguous -> one aligned v8f store each.
// DOLOG: apply logf() in the epilogue for blockIdx.z == 0 only.
template<bool TRANS_STORE, bool DOLOG>
__global__ void gemm_nt_wmma(const float* __restrict__ A0,
                             const float* __restrict__ A1,
                             const float* __restrict__ B0,
                             const float* __restrict__ B1,
                             float* __restrict__ C0,
                             float* __restrict__ C1,
                             int M, int N, int K) {
    const float* A = blockIdx.z ? A1 : A0;
    const float* Bt = blockIdx.z ? B1 : B0;
    float*       C = blockIdx.z ? C1 : C0;
    const bool dolog = DOLOG && (blockIdx.z == 0);

    int lane = threadIdx.x & 31;
    int wave = threadIdx.x >> 5;
    int lh   = lane >> 4;       // half-wave: selects K pair {0,1} vs {2,3}
    int l16  = lane & 15;       // M (for A) / N (for B,C) index within 16-tile
    int row0 = blockIdx.y * 256 + wave * 32;
    int col0 = blockIdx.x * 64;

    v8f acc[2][4] = {};
    const float* arow0 = A + (long long)(row0 +      l16) * K;
    const float* arow1 = A + (long long)(row0 + 16 + l16) * K;

    #pragma unroll 2
    for (int k = 0; k < K; k += 4) {
        int ka = k + lh * 2;
        v2f a0 = *(const v2f*)(arow0 + ka);         // A[m,      ka..ka+1]
        v2f a1 = *(const v2f*)(arow1 + ka);         // A[m + 16, ka..ka+1]
        #pragma unroll
        for (int n = 0; n < 4; ++n) {
            const float* bp = Bt + (long long)(col0 + n * 16 + l16) * K + ka;
            v2f b = *(const v2f*)bp;                // Bt[n, ka..ka+1] (contiguous)
            acc[0][n] = wmma_f32_k4(a0, b, acc[0][n]);
            acc[1][n] = wmma_f32_k4(a1, b, acc[1][n]);
        }
    }

    #pragma unroll
    for (int m = 0; m < 2; ++m) {
        #pragma unroll
        for (int n = 0; n < 4; ++n) {
            int cc = col0 + n * 16 + l16;
            if (TRANS_STORE) {
                // Ct[cc, row0 + m*16 + lh*8 .. +8] : 8 M-contiguous floats
                *(v8f*)(C + (long long)cc * M + row0 + m * 16 + lh * 8) = acc[m][n];
            } else {
                #pragma unroll
                for (int r = 0; r < 8; ++r) {
                    int rr = row0 + m * 16 + r + lh * 8;
                    float v = acc[m][n][r];
                    C[(long long)rr * N + cc] = dolog ? logf(v) : v;
                }
            }
        }
    }
}

extern "C" void kernel_launch(void* const* d_in, const int* in_sizes, int n_in,
                              void* d_out, int out_size, void* d_ws, size_t ws_size,
                              hipStream_t stream) {
    const int*   user_idx     = (const int*)d_in[0];
    const int*   item_idx     = (const int*)d_in[1];
    const float* user_mixture = (const float*)d_in[2];
    const float* item_mixture = (const float*)d_in[3];
    const float* gaussian_1   = (const float*)d_in[4];
    const float* gaussian_2   = (const float*)d_in[5];

    // Output layout (flat, return order): kl[B*B], transition[B*B], m1[B*MU], m2[B*MI]
    float* out = (float*)d_out;
    float* kl  = out;
    float* tr  = out + (long long)BSZ * BSZ;
    float* m1  = tr  + (long long)BSZ * BSZ;
    float* m2  = m1  + (long long)BSZ * MUC;

    // Workspace: gm[512*512] | tp[512*512] | T1t[4096*512] | T2t[4096*512] (18 MB)
    // T1t/T2t hold (gm@m2^T)^T and (tp@m2^T)^T, i.e. [B rows, MU cols].
    float* gm  = (float*)d_ws;
    float* tp  = gm  + MUC * MIC;
    float* T1t = tp  + MUC * MIC;
    float* T2t = T1t + (long long)MUC * BSZ;

    // 1) gathers
    gather_rows<<<BSZ, 128, 0, stream>>>(user_idx, user_mixture, m1, MUC);
    gather_rows<<<BSZ, 128, 0, stream>>>(item_idx, item_mixture, m2, MIC);

    // 2) Gaussian expected likelihood matrix
    gm_kernel<<<(MUC * MIC) / 256, 256, 0, stream>>>(gaussian_1, gaussian_2, gm);

    // 3) row softmax of -gm
    softmax_kernel<<<MUC, 256, 0, stream>>>(gm, tp);

    // 4) T1t = (gm @ m2^T)^T ; T2t = (tp @ m2^T)^T   (M=512, N=4096, K=512)
    //    Transposed store -> stage 5 becomes NT with contiguous B loads.
    dim3 g4(BSZ / 64, MUC / 256, 2);
    gemm_nt_wmma<true, false><<<g4, 256, 0, stream>>>(
        gm, tp, m2, m2, T1t, T2t, MUC, BSZ, MIC);

    // 5) kl = log(m1 @ T1t^T) ; tr = m1 @ T2t^T     (M=4096, N=4096, K=512)
    dim3 g5(BSZ / 64, BSZ / 256, 2);
    gemm_nt_wmma<false, true><<<g5, 256, 0, stream>>>(
        m1, m1, T1t, T2t, kl, tr, BSZ, BSZ, MUC);
}